// BGFAD_64991445123099
// MI455X (gfx1250) — compile-verified
//
#include <hip/hip_runtime.h>
#include <math.h>

#define NPTS   4096
#define NBATCH 8
#define BIGF   1.0e10f
#define SENT   3.0e38f
#define R1SQ   0.0064f      // 0.08^2
#define R2SQ   0.0625f      // 0.25^2

typedef __attribute__((ext_vector_type(2))) float v2f;
typedef __attribute__((ext_vector_type(8))) float v8f;

// ---------- async global->LDS staging (CDNA5 ASYNCcnt path) ----------
#if defined(__has_builtin)
#  if __has_builtin(__builtin_amdgcn_global_load_async_to_lds_b128) && \
      __has_builtin(__builtin_amdgcn_s_wait_asynccnt)
#    define USE_ASYNC_LDS 1
#  endif
#endif
#ifndef USE_ASYNC_LDS
#  define USE_ASYNC_LDS 0
#endif

#if USE_ASYNC_LDS
// exact pointee type from the builtin's signature:
//   "__attribute__((__vector_size__(4 * sizeof(int)))) int __device__ *"
typedef int v4i_vs __attribute__((vector_size(4 * sizeof(int))));
typedef __attribute__((address_space(1))) v4i_vs* gvec_p;   // global (AS1)
typedef __attribute__((address_space(3))) v4i_vs* lvec_p;   // LDS (AS3)
#endif

__device__ __forceinline__ void stage_x_to_lds(const float* __restrict__ gsrc,
                                               float* lds_dst, int tid) {
#if USE_ASYNC_LDS
  for (int i = tid * 4; i < NPTS * 3; i += 256 * 4) {
    __builtin_amdgcn_global_load_async_to_lds_b128(
        (gvec_p)(gsrc + i), (lvec_p)(lds_dst + i), 0, 0);
  }
  __builtin_amdgcn_s_wait_asynccnt(0);   // this wave's copies done before barrier
#else
  for (int i = tid; i < NPTS * 3; i += 256) lds_dst[i] = gsrc[i];
#endif
}

// ---------- small branchless sorted-list helpers ----------
__device__ __forceinline__ void cswapF(float &a, float &b) {
  float lo = fminf(a, b);
  float hi = fmaxf(a, b);
  a = lo; b = hi;
}

// maintain 4 smallest floats (sorted s0<=s1<=s2<=s3)
__device__ __forceinline__ void insert4F(float d, float &s0, float &s1,
                                         float &s2, float &s3) {
  s3 = fminf(s3, d);
  cswapF(s2, s3);
  cswapF(s1, s2);
  cswapF(s0, s1);
}

__device__ __forceinline__ bool lessKV(float da, int ia, float db, int ib) {
  return (da < db) || ((da == db) && (ia < ib));  // top_k lowest-index tiebreak
}

__device__ __forceinline__ void cswapKV(float &da, int &ia, float &db, int &ib) {
  bool sw = lessKV(db, ib, da, ia);
  float td = sw ? db : da; int ti = sw ? ib : ia;
  float bd = sw ? da : db; int bi = sw ? ia : ib;
  da = td; ia = ti; db = bd; ib = bi;
}

__device__ __forceinline__ void insert4(float d, int j,
    float &d0, int &i0, float &d1, int &i1,
    float &d2, int &i2, float &d3, int &i3) {
  bool sw = lessKV(d, j, d3, i3);
  d3 = sw ? d : d3; i3 = sw ? j : i3;
  cswapKV(d2, i2, d3, i3);
  cswapKV(d1, i1, d2, i2);
  cswapKV(d0, i0, d1, i1);
}

// =========================================================
// K1: SOR pass — top-4 smallest sq-dists, drop first, mean rest
// grid (32, NBATCH), block 256 (8 wave32 / WG, 16 queries / wave)
// =========================================================
__global__ __launch_bounds__(256) void k_sor(const float* __restrict__ x,
                                             float* __restrict__ vout) {
  const int b = blockIdx.y;
  const float* xb = x + (size_t)b * NPTS * 3;
  __shared__ alignas(16) float sx[NPTS * 3];   // raw coords (48 KB)
  __shared__ float sxx[NPTS];                  // squared norms (16 KB)
  const int tid = threadIdx.x;
  __builtin_prefetch(xb + tid * 48, 0, 0);     // global_prefetch_b8
  stage_x_to_lds(xb, sx, tid);
  __syncthreads();
  for (int i = tid; i < NPTS; i += 256) {
    float px = sx[i * 3 + 0], py = sx[i * 3 + 1], pz = sx[i * 3 + 2];
    sxx[i] = px * px + py * py + pz * pz;
  }
  __syncthreads();

  const int wave = tid >> 5;
  const int lane = tid & 31;
  const int l16  = lane & 15;
  const bool hi  = lane >= 16;
  const int q = blockIdx.x * 128 + wave * 16 + l16;   // query = column N
  v2f bv;                        // B 4x16: lanes0-15 K=0,1 ; lanes16-31 K=2,3
  bv.x = hi ? sx[q * 3 + 2] : sx[q * 3 + 0];
  bv.y = hi ? 1.0f : sx[q * 3 + 1];
  const float xxq = sxx[q];
  const int off = hi ? 8 : 0;    // C rows: M = r + off

  float s0 = SENT, s1 = SENT, s2 = SENT, s3 = SENT;
  for (int jb = 0; jb < NPTS; jb += 16) {
    const int m = jb + l16;      // A row M = l16 (both half-lanes)
    v2f av;                      // A = (-2x, -2y | -2z, ||x||^2)
    av.x = hi ? (-2.0f * sx[m * 3 + 2]) : (-2.0f * sx[m * 3 + 0]);
    av.y = hi ? sxx[m] : (-2.0f * sx[m * 3 + 1]);
    v8f c = {0.f, 0.f, 0.f, 0.f, 0.f, 0.f, 0.f, 0.f};
    c = __builtin_amdgcn_wmma_f32_16x16x4_f32(false, av, false, bv,
                                              (short)0, c, false, false);
#pragma unroll
    for (int r = 0; r < 8; ++r) {
      float d = fmaxf(c[r] + xxq, 0.0f);  // ||xj||^2 - 2<xj,xq> + ||xq||^2
      insert4F(d, s0, s1, s2, s3);        // self stays in; dropped as s0 below
    }
  }
  // merge the two half-lanes holding the same query column
  float p0 = __shfl_xor(s0, 16, 32);
  float p1 = __shfl_xor(s1, 16, 32);
  float p2 = __shfl_xor(s2, 16, 32);
  float p3 = __shfl_xor(s3, 16, 32);
  insert4F(p0, s0, s1, s2, s3);
  insert4F(p1, s0, s1, s2, s3);
  insert4F(p2, s0, s1, s2, s3);
  insert4F(p3, s0, s1, s2, s3);
  // reference: mean(top_k(-d,4)[1:]) == drop smallest, average rest
  if (!hi) vout[b * NPTS + q] = (s1 + s2 + s3) * (1.0f / 3.0f);
}

// =========================================================
// K2: per-batch mean/std(ddof=1) of v  ->  mask1
// grid (NBATCH), block 256
// =========================================================
__global__ __launch_bounds__(256) void k_stats(const float* __restrict__ v,
                                               int* __restrict__ mask1) {
  const int b = blockIdx.x;
  const float* vb = v + b * NPTS;
  __shared__ float ssum[256];
  __shared__ float ssq[256];
  const int tid = threadIdx.x;
  float s = 0.f, s2 = 0.f;
  for (int i = tid; i < NPTS; i += 256) {
    float t = vb[i];
    s += t; s2 += t * t;
  }
  ssum[tid] = s; ssq[tid] = s2;
  __syncthreads();
  for (int stp = 128; stp > 0; stp >>= 1) {
    if (tid < stp) { ssum[tid] += ssum[tid + stp]; ssq[tid] += ssq[tid + stp]; }
    __syncthreads();
  }
  const float mean = ssum[0] * (1.0f / NPTS);
  const float var  = (ssq[0] - (float)NPTS * mean * mean) * (1.0f / (NPTS - 1));
  const float sd   = sqrtf(fmaxf(var, 0.f));
  const float lo = mean - 0.15f * sd;
  const float hiT = mean + 1.1f * sd;      // ALPHA_SOR
  for (int i = tid; i < NPTS; i += 256) {
    float t = vb[i];
    mask1[b * NPTS + i] = (t > lo && t < hiT) ? 1 : 0;
  }
}

// =========================================================
// K3: masked 4-NN with indices (reproduces top_k(-d1, 4)[1:])
// grid (32, NBATCH), block 256
// =========================================================
__global__ __launch_bounds__(256) void k_knn(const float* __restrict__ x,
                                             const int* __restrict__ mask1,
                                             int* __restrict__ nbidx) {
  const int b = blockIdx.y;
  const float* xb = x + (size_t)b * NPTS * 3;
  const int* mb = mask1 + b * NPTS;
  __shared__ alignas(16) float sx[NPTS * 3];
  __shared__ float sxx[NPTS];
  __shared__ alignas(8) unsigned char sm[NPTS];
  const int tid = threadIdx.x;
  stage_x_to_lds(xb, sx, tid);
  for (int i = tid; i < NPTS; i += 256) sm[i] = (unsigned char)mb[i];
  __syncthreads();
  for (int i = tid; i < NPTS; i += 256) {
    float px = sx[i * 3 + 0], py = sx[i * 3 + 1], pz = sx[i * 3 + 2];
    sxx[i] = px * px + py * py + pz * pz;
  }
  __syncthreads();

  const int wave = tid >> 5;
  const int lane = tid & 31;
  const int l16  = lane & 15;
  const bool hi  = lane >= 16;
  const int q = blockIdx.x * 128 + wave * 16 + l16;
  v2f bv;
  bv.x = hi ? sx[q * 3 + 2] : sx[q * 3 + 0];
  bv.y = hi ? 1.0f : sx[q * 3 + 1];
  const float xxq = sxx[q];
  const int off = hi ? 8 : 0;
  const bool mq = sm[q] != 0;

  float d0 = SENT, d1 = SENT, d2 = SENT, d3 = SENT;
  int   i0 = -1,   i1 = -1,   i2 = -1,   i3 = -1;
  for (int jb = 0; jb < NPTS; jb += 16) {
    const int m = jb + l16;
    v2f av;
    av.x = hi ? (-2.0f * sx[m * 3 + 2]) : (-2.0f * sx[m * 3 + 0]);
    av.y = hi ? sxx[m] : (-2.0f * sx[m * 3 + 1]);
    v8f c = {0.f, 0.f, 0.f, 0.f, 0.f, 0.f, 0.f, 0.f};
    c = __builtin_amdgcn_wmma_f32_16x16x4_f32(false, av, false, bv,
                                              (short)0, c, false, false);
    // 8 neighbor-mask bytes in one LDS read (jb+off is 8B aligned)
    const unsigned long long mb8 =
        *(const unsigned long long*)(sm + jb + off);
#pragma unroll
    for (int r = 0; r < 8; ++r) {
      int j = jb + off + r;
      float d = fmaxf(c[r] + xxq, 0.0f);
      bool valid = mq && (((mb8 >> (8 * r)) & 1ULL) != 0);
      float dd = valid ? d : BIGF;
      insert4(dd, j, d0, i0, d1, i1, d2, i2, d3, i3);
    }
  }
  // cross-half merge (index tiebreak matters here)
  float pd0 = __shfl_xor(d0, 16, 32); int pi0 = __shfl_xor(i0, 16, 32);
  float pd1 = __shfl_xor(d1, 16, 32); int pi1 = __shfl_xor(i1, 16, 32);
  float pd2 = __shfl_xor(d2, 16, 32); int pi2 = __shfl_xor(i2, 16, 32);
  float pd3 = __shfl_xor(d3, 16, 32); int pi3 = __shfl_xor(i3, 16, 32);
  insert4(pd0, pi0, d0, i0, d1, i1, d2, i2, d3, i3);
  insert4(pd1, pi1, d0, i0, d1, i1, d2, i2, d3, i3);
  insert4(pd2, pi2, d0, i0, d1, i1, d2, i2, d3, i3);
  insert4(pd3, pi3, d0, i0, d1, i1, d2, i2, d3, i3);
  if (!hi) {
    int base = (b * NPTS + q) * 3;     // drop idx[0]
    nbidx[base + 0] = i1;
    nbidx[base + 1] = i2;
    nbidx[base + 2] = i3;
  }
}

// =========================================================
// K4: per-point 3x3 covariance + closed-form eigvals -> curvature
// grid (B*N/256), block 256
// =========================================================
__global__ __launch_bounds__(256) void k_curv(const float* __restrict__ x,
                                              const int* __restrict__ nbidx,
                                              float* __restrict__ curv) {
  const int gid = blockIdx.x * 256 + threadIdx.x;       // 0..B*N-1
  const int b = gid >> 12;
  const float* xb = x + (size_t)b * NPTS * 3;
  const int* nb = nbidx + (size_t)gid * 3;
  float px[3], py[3], pz[3];
#pragma unroll
  for (int k = 0; k < 3; ++k) {
    int j = nb[k];
    px[k] = xb[j * 3 + 0];
    py[k] = xb[j * 3 + 1];
    pz[k] = xb[j * 3 + 2];
  }
  const float mx = (px[0] + px[1] + px[2]) * (1.0f / 3.0f);
  const float my = (py[0] + py[1] + py[2]) * (1.0f / 3.0f);
  const float mz = (pz[0] + pz[1] + pz[2]) * (1.0f / 3.0f);
  float cxx = 0, cxy = 0, cxz = 0, cyy = 0, cyz = 0, czz = 0;
#pragma unroll
  for (int k = 0; k < 3; ++k) {
    float dx = px[k] - mx, dy = py[k] - my, dz = pz[k] - mz;
    cxx += dx * dx; cxy += dx * dy; cxz += dx * dz;
    cyy += dy * dy; cyz += dy * dz; czz += dz * dz;
  }
  cxx *= 0.5f; cxy *= 0.5f; cxz *= 0.5f;     // / (K_CUR - 1)
  cyy *= 0.5f; cyz *= 0.5f; czz *= 0.5f;
  const float tr = cxx + cyy + czz;
  const float qm = tr * (1.0f / 3.0f);
  const float p1 = cxy * cxy + cxz * cxz + cyz * cyz;
  const float aa = cxx - qm, bb = cyy - qm, cc = czz - qm;
  const float p2 = aa * aa + bb * bb + cc * cc + 2.0f * p1;
  float lmin;
  if (p2 <= 1e-30f) {
    lmin = qm;
  } else {
    const float p = sqrtf(p2 * (1.0f / 6.0f));
    const float inv = 1.0f / p;
    const float b00 = aa * inv, b11 = bb * inv, b22 = cc * inv;
    const float b01 = cxy * inv, b02 = cxz * inv, b12 = cyz * inv;
    float detB = b00 * (b11 * b22 - b12 * b12)
               - b01 * (b01 * b22 - b12 * b02)
               + b02 * (b01 * b12 - b11 * b02);
    float r = fminf(1.0f, fmaxf(-1.0f, 0.5f * detB));
    float phi = acosf(r) * (1.0f / 3.0f);
    lmin = qm + 2.0f * p * cosf(phi + 2.0943951023931953f);  // smallest eig
  }
  curv[gid] = lmin / (tr + 1e-6f);
}

// =========================================================
// K5: neighbor-curvature stats (ddof=0) -> mask2
// =========================================================
__global__ __launch_bounds__(256) void k_mask2(const float* __restrict__ curv,
                                               const int* __restrict__ nbidx,
                                               const int* __restrict__ mask1,
                                               int* __restrict__ mask2) {
  const int gid = blockIdx.x * 256 + threadIdx.x;
  const float* cb = curv + (gid & ~(NPTS - 1));   // batch base
  const int* nb = nbidx + (size_t)gid * 3;
  const float c0 = cb[nb[0]], c1 = cb[nb[1]], c2 = cb[nb[2]];
  const float mc = (c0 + c1 + c2) * (1.0f / 3.0f);
  const float v0 = c0 - mc, v1 = c1 - mc, v2 = c2 - mc;
  const float sc = sqrtf((v0 * v0 + v1 * v1 + v2 * v2) * (1.0f / 3.0f));
  const float cu = curv[gid];
  const int m = (mask1[gid] != 0) &&
                (cu >= mc - 1.8f * sc) && (cu <= mc + 1.8f * sc);
  mask2[gid] = m;
}

// =========================================================
// K6: radius counting + final outputs (pts_out, mask3)
// grid (32, NBATCH), block 256
// =========================================================
__global__ __launch_bounds__(256) void k_count(const float* __restrict__ x,
                                               const int* __restrict__ mask2,
                                               float* __restrict__ out) {
  const int b = blockIdx.y;
  const float* xb = x + (size_t)b * NPTS * 3;
  const int* mb = mask2 + b * NPTS;
  __shared__ alignas(16) float sx[NPTS * 3];
  __shared__ float sxx[NPTS];
  __shared__ alignas(8) unsigned char sm[NPTS];
  const int tid = threadIdx.x;
  stage_x_to_lds(xb, sx, tid);
  for (int i = tid; i < NPTS; i += 256) sm[i] = (unsigned char)mb[i];
  __syncthreads();
  for (int i = tid; i < NPTS; i += 256) {
    float px = sx[i * 3 + 0], py = sx[i * 3 + 1], pz = sx[i * 3 + 2];
    sxx[i] = px * px + py * py + pz * pz;
  }
  __syncthreads();

  const int wave = tid >> 5;
  const int lane = tid & 31;
  const int l16  = lane & 15;
  const bool hi  = lane >= 16;
  const int q = blockIdx.x * 128 + wave * 16 + l16;
  v2f bv;
  bv.x = hi ? sx[q * 3 + 2] : sx[q * 3 + 0];
  bv.y = hi ? 1.0f : sx[q * 3 + 1];
  const float xxq = sxx[q];
  const int off = hi ? 8 : 0;

  int c1 = 0, c2 = 0;
  for (int jb = 0; jb < NPTS; jb += 16) {
    const int m = jb + l16;
    v2f av;
    av.x = hi ? (-2.0f * sx[m * 3 + 2]) : (-2.0f * sx[m * 3 + 0]);
    av.y = hi ? sxx[m] : (-2.0f * sx[m * 3 + 1]);
    v8f c = {0.f, 0.f, 0.f, 0.f, 0.f, 0.f, 0.f, 0.f};
    c = __builtin_amdgcn_wmma_f32_16x16x4_f32(false, av, false, bv,
                                              (short)0, c, false, false);
    const unsigned long long mb8 =
        *(const unsigned long long*)(sm + jb + off);
#pragma unroll
    for (int r = 0; r < 8; ++r) {
      float d = fmaxf(c[r] + xxq, 0.0f);
      int mj = (int)((mb8 >> (8 * r)) & 1ULL);
      c1 += (d <= R1SQ) ? mj : 0;
      c2 += (d <= R2SQ) ? mj : 0;
    }
  }
  c1 += __shfl_xor(c1, 16, 32);
  c2 += __shfl_xor(c2, 16, 32);
  const int m3 = (sm[q] && (c1 >= 3) && (c2 > 20)) ? 1 : 0;
  if (!hi) {
    const float f = (float)m3;
    const size_t gq = (size_t)b * NPTS + q;
    out[gq * 3 + 0] = sx[q * 3 + 0] * f;
    out[gq * 3 + 1] = sx[q * 3 + 1] * f;
    out[gq * 3 + 2] = sx[q * 3 + 2] * f;
    out[(size_t)NBATCH * NPTS * 3 + gq] = f;   // mask3 as float
  }
}

// =========================================================
extern "C" void kernel_launch(void* const* d_in, const int* in_sizes, int n_in,
                              void* d_out, int out_size, void* d_ws, size_t ws_size,
                              hipStream_t stream) {
  (void)in_sizes; (void)n_in; (void)out_size; (void)ws_size;
  const float* x = (const float*)d_in[0];
  char* ws = (char*)d_ws;
  float* v     = (float*)(ws + 0);        // 131072 B
  int*   mask1 = (int*)  (ws + 131072);   // 131072 B
  int*   nbidx = (int*)  (ws + 262144);   // 393216 B
  float* curv  = (float*)(ws + 655360);   // 131072 B
  int*   mask2 = (int*)  (ws + 786432);   // 131072 B
  float* out = (float*)d_out;

  dim3 blk(256);
  dim3 gdist(32, NBATCH);
  k_sor  <<<gdist,      blk, 0, stream>>>(x, v);
  k_stats<<<dim3(NBATCH), blk, 0, stream>>>(v, mask1);
  k_knn  <<<gdist,      blk, 0, stream>>>(x, mask1, nbidx);
  k_curv <<<dim3(NBATCH * NPTS / 256), blk, 0, stream>>>(x, nbidx, curv);
  k_mask2<<<dim3(NBATCH * NPTS / 256), blk, 0, stream>>>(curv, nbidx, mask1, mask2);
  k_count<<<gdist,      blk, 0, stream>>>(x, mask2, out);
}